// MGEVelocity_71244917506142
// MI455X (gfx1250) — compile-verified
//
#include <hip/hip_runtime.h>
#include <math.h>
#include <stdint.h>

#define KC   16        // N_COMP
#define MAXQ 256       // supports quad_points up to 256 (reference uses 128)
#define G_CONST 0.004301f

typedef unsigned int v4u __attribute__((ext_vector_type(4)));
typedef int          v8i __attribute__((ext_vector_type(8)));
typedef int          v4i __attribute__((ext_vector_type(4)));

// ---------------------------------------------------------------------------
// Setup kernel: one block. Builds the fused (C*log2e, W) table (Q*16 float2)
// and 4 scalars into d_ws. Gauss-Legendre nodes via double-precision Newton.
// ---------------------------------------------------------------------------
__global__ __launch_bounds__(256) void mge_setup(
    const float* __restrict__ surf, const float* __restrict__ sigma,
    const float* __restrict__ qobs, const float* __restrict__ M_to_L,
    const float* __restrict__ inc,  const float* __restrict__ m_bh,
    const int*  __restrict__ qp,
    float2* __restrict__ cw, float* __restrict__ scal)
{
    __shared__ float  s_coef[KC], s_invsig2[KC], s_q2[KC];
    __shared__ float  s_scale;
    __shared__ double s_half, s_mid;
    __shared__ int    s_Q;

    const int tid = threadIdx.x;

    if (tid == 0) {
        int Q = qp[0];
        if (Q > MAXQ) Q = MAXQ;
        s_Q = Q;

        // median(sigma) via insertion sort of 16 values; quantile(0.5) with
        // linear interpolation at position 7.5 -> (s[7]+s[8])/2
        float s[KC];
        for (int i = 0; i < KC; ++i) s[i] = sigma[i];
        for (int i = 1; i < KC; ++i) {
            float v = s[i]; int j = i - 1;
            while (j >= 0 && s[j] > v) { s[j+1] = s[j]; --j; }
            s[j+1] = v;
        }
        float scale = 0.5f * (s[7] + s[8]);
        s_scale = scale;
        float mds = 0.5f * (s[7]/scale + s[8]/scale);   // ~1.0
        float mxs = s[KC-1] / scale;

        double t_low  = asinh(log(1e-7  * (double)mds) * 2.0 / M_PI);
        double t_high = asinh(log(1000.0 * (double)mxs) * 2.0 / M_PI);
        s_half = 0.5 * (t_high - t_low);
        s_mid  = 0.5 * (t_high + t_low);

        scal[0] = (float)(2.0 * M_PI) * G_CONST * scale * scale;   // mge coef
        scal[1] = G_CONST * exp10f(m_bh[0]) / scale;               // bh coef
        scal[2] = 1.0f / (scale * scale);                          // 1/scale^2
        scal[3] = (float)(Q * KC);                                 // term count
    }
    __syncthreads();

    const int Q = s_Q;

    if (tid < KC) {
        float ci = cosf(inc[0]), si = sinf(inc[0]);
        float qo = qobs[tid];
        float qi = sqrtf(qo*qo - ci*ci) / si;
        float sg = sigma[tid];
        float md = surf[tid] * M_to_L[0] * qo / (qi * sg * sqrtf(2.0f * (float)M_PI));
        s_coef[tid]    = qi * md;
        float ssc      = sg / s_scale;
        s_invsig2[tid] = 1.0f / (ssc * ssc);
        s_q2[tid]      = qi * qi;
    }
    __syncthreads();

    if (tid < Q) {
        // Gauss-Legendre node #tid of order Q: Newton iteration in double.
        const int n = Q;
        double x0 = cos(M_PI * ((double)tid + 0.75) / ((double)n + 0.5));
        double dp = 1.0;
        for (int it = 0; it < 64; ++it) {
            double pa = 1.0, pb = x0;
            for (int k = 2; k <= n; ++k) {
                double pc = ((2.0*k - 1.0) * x0 * pb - (k - 1.0) * pa) / (double)k;
                pa = pb; pb = pc;
            }
            dp = (double)n * (x0 * pb - pa) / (x0*x0 - 1.0);
            double dx = pb / dp;
            x0 -= dx;
            if (fabs(dx) < 1e-15) break;
        }
        double w = 2.0 / ((1.0 - x0*x0) * dp * dp);

        // double-exponential transform
        double t    = s_half * x0 + s_mid;
        double u    = exp(0.5 * M_PI * sinh(t));
        double duw  = 0.5 * M_PI * cosh(t) * u * (s_half * w);
        double onep = 1.0 + u;
        double aj   = -0.5 / onep;
        const double LOG2E = 1.4426950408889634;

        for (int k = 0; k < KC; ++k) {
            double denom = onep * onep * sqrt((double)s_q2[k] + u);
            float2 p;
            p.x = (float)(aj * (double)s_invsig2[k] * LOG2E);  // exp2 coefficient
            p.y = (float)(duw * (double)s_coef[k] / denom);    // weight
            cw[tid * KC + k] = p;
        }
    }
}

// ---------------------------------------------------------------------------
// Main kernel: TDM-stage the 16KB (C,W) table into LDS, then per point:
// 2048 x { ds_load_b64, v_mul, v_exp_f32, v_fmac } -> vc2 -> v_rot.
// ---------------------------------------------------------------------------
__global__ __launch_bounds__(256) void mge_main(
    const float* __restrict__ xg, const float* __restrict__ yg,
    const float* __restrict__ zg,
    const float* __restrict__ cwg, const float* __restrict__ scal,
    float* __restrict__ out, int N)
{
    __shared__ float2 cw[MAXQ * KC];

    const float mge_coef = scal[0];
    const float bh_coef  = scal[1];
    const float inv_s2   = scal[2];
    const int   terms    = __builtin_amdgcn_readfirstlane((int)scal[3]);

    // Wave 0 issues one Tensor-Data-Mover load of the whole table into LDS.
    if (threadIdx.x < 32u) {
        uint64_t ga  = (uint64_t)(uintptr_t)cwg;
        uint32_t lds = (uint32_t)(uintptr_t)(void*)&cw[0];   // flat low 32 = LDS byte addr
        uint32_t ndw = (uint32_t)terms * 2u;                 // dwords to move

        v4u g0;
        g0[0] = 1u;                                              // count=1, user descriptor
        g0[1] = lds;                                             // lds_addr
        g0[2] = (uint32_t)ga;                                    // global_addr[31:0]
        g0[3] = (uint32_t)((ga >> 32) & 0x1FFFFFFu) | 0x80000000u; // addr[56:32] | type=2

        v8i g1;
        g1[0] = (int)(2u << 16);                   // data_size = 4B
        g1[1] = (int)((ndw & 0xFFFFu) << 16);      // tensor_dim0[15:0]
        g1[2] = (int)(((ndw >> 16) & 0xFFFFu) | (1u << 16)); // dim0 hi | tensor_dim1=1
        g1[3] = (int)((ndw & 0xFFFFu) << 16);      // tile_dim0 = ndw
        g1[4] = 0;                                 // tile_dim1/2 unused
        g1[5] = (int)ndw;                          // tensor_dim0_stride
        g1[6] = 0;
        g1[7] = 0;

        v4i g2 = {0, 0, 0, 0};                     // group 2: unused (<=2D)
        v4i g3 = {0, 0, 0, 0};                     // group 3: unused (<=2D)
        v8i g4 = {0, 0, 0, 0, 0, 0, 0, 0};         // extra arg (clang-23 form): zero-fill
        __builtin_amdgcn_tensor_load_to_lds(g0, g1, g2, g3, g4, 0);
        __builtin_amdgcn_s_wait_tensorcnt(0);
    }
    __syncthreads();

    const int i = blockIdx.x * blockDim.x + threadIdx.x;
    if (i < N) {
        float xv = xg[i], yv = yg[i], zv = zg[i];
        float R2 = fmaf(xv, xv, fmaf(yv, yv, zv * zv)) * inv_s2;   // (R/scale)^2

        float a0 = 0.f, a1 = 0.f, a2 = 0.f, a3 = 0.f;
        int t = 0;
        for (; t + 4 <= terms; t += 4) {
            float2 p0 = cw[t+0], p1 = cw[t+1], p2 = cw[t+2], p3 = cw[t+3];
            a0 = fmaf(p0.y, __builtin_amdgcn_exp2f(p0.x * R2), a0);
            a1 = fmaf(p1.y, __builtin_amdgcn_exp2f(p1.x * R2), a1);
            a2 = fmaf(p2.y, __builtin_amdgcn_exp2f(p2.x * R2), a2);
            a3 = fmaf(p3.y, __builtin_amdgcn_exp2f(p3.x * R2), a3);
        }
        for (; t < terms; ++t) {
            float2 p = cw[t];
            a0 = fmaf(p.y, __builtin_amdgcn_exp2f(p.x * R2), a0);
        }
        float acc = (a0 + a1) + (a2 + a3);

        float r   = sqrtf(R2);                      // R_sc
        float vc2 = fmaf(mge_coef, acc, bh_coef / (R2 * r));
        out[i]    = r * sqrtf(vc2);
    }
}

// ---------------------------------------------------------------------------
extern "C" void kernel_launch(void* const* d_in, const int* in_sizes, int n_in,
                              void* d_out, int out_size, void* d_ws, size_t ws_size,
                              hipStream_t stream)
{
    const float* x      = (const float*)d_in[0];
    const float* y      = (const float*)d_in[1];
    const float* z      = (const float*)d_in[2];
    const float* surf   = (const float*)d_in[3];
    const float* sigma  = (const float*)d_in[4];
    const float* qobs   = (const float*)d_in[5];
    const float* M_to_L = (const float*)d_in[6];
    const float* inc    = (const float*)d_in[7];
    const float* m_bh   = (const float*)d_in[8];
    const int*   qp     = (const int*)d_in[9];

    float*  ws   = (float*)d_ws;
    float2* cw   = (float2*)ws;                  // [MAXQ*KC] float2 table
    float*  scal = ws + MAXQ * KC * 2;           // scalars after the table

    const int N = in_sizes[0];

    mge_setup<<<1, 256, 0, stream>>>(surf, sigma, qobs, M_to_L, inc, m_bh,
                                     qp, cw, scal);
    mge_main<<<(N + 255) / 256, 256, 0, stream>>>(x, y, z, (const float*)cw,
                                                  scal, (float*)d_out, N);
}